// MY_GCLSTM__41901700940311
// MI455X (gfx1250) — compile-verified
//
#include <hip/hip_runtime.h>
#include <math.h>

// GCLSTM (ChebConv K=2) fused cell + head for MI455X / gfx1250.
// Edge aggregation (bandwidth/atomic bound) in dedicated kernels; all dense
// math via V_WMMA_F32_16X16X4_F32 with weights staged in LDS in K-pair-major
// layout so each WMMA B-operand is a single ds_load_b64.

#define NODES 100000
#define EDGES 1600000
#define FIN   64
#define FOUT  32

typedef __attribute__((ext_vector_type(2))) float v2f;
typedef __attribute__((ext_vector_type(8))) float v8f;

// D = A(16x4 f32) * B(4x16 f32) + C(16x16 f32)
__device__ __forceinline__ v8f wmma_k4(v2f a, v2f b, v8f c) {
  return __builtin_amdgcn_wmma_f32_16x16x4_f32(
      /*neg_a=*/false, a, /*neg_b=*/false, b,
      /*c_mod=*/(short)0, c, /*reuse_a=*/false, /*reuse_b=*/false);
}

__device__ __forceinline__ float sigmoid_(float x) {
  return 1.0f / (1.0f + __expf(-x));
}

// ---------------------------------------------------------------- degree ----
__global__ void k_deg(const int* __restrict__ row, const float* __restrict__ ew,
                      float* __restrict__ deg) {
  int e = blockIdx.x * 256 + threadIdx.x;
  if (e < EDGES) atomicAdd(&deg[row[e]], ew[e]);
}

// deg -> d^{-1/2} in place (0 where deg<=0)
__global__ void k_dis(float* __restrict__ deg) {
  int n = blockIdx.x * 256 + threadIdx.x;
  if (n < NODES) {
    float d = deg[n];
    deg[n] = d > 0.0f ? rsqrtf(d) : 0.0f;
  }
}

// ------------------------------------------------- edge scatter (agg) -------
// One wave32 per edge; lane = feature. agg[row] += w_norm * h[col].
__global__ void k_scatter(const int* __restrict__ row, const int* __restrict__ col,
                          const float* __restrict__ ew, const float* __restrict__ dis,
                          const float* __restrict__ h, float* __restrict__ agg) {
  int wid  = blockIdx.x * 8 + (threadIdx.x >> 5);   // edge id
  int lane = threadIdx.x & 31;                      // feature id
  if (wid < EDGES) {
    int r = row[wid];
    int c = col[wid];
    float w = dis[r] * ew[wid] * dis[c];            // symmetric normalization
    atomicAdd(&agg[r * FOUT + lane], w * h[c * FOUT + lane]);
  }
}

// ----------------------------------------------------- fused gate kernel ----
// Block = 256 threads = 8 waves; each wave owns a 16-node tile.
// LDS (floats), all B-matrices stored K-pair-major:
//   element (k, col) of a 32-col matrix lives at (k>>1)*64 + col*2 + (k&1),
//   so the WMMA B operand (B[k][col], B[k+1][col]) is one aligned b64.
// Layout: W[4][2048] @0 | th0[4][1024] @8192 | th1[4][1024] @12288
//         | b[4][32] @16384 | cb[4][32] @16512 | wlin[32] @16640 | blin @16672
__global__ void __launch_bounds__(256, 1)
k_gclstm_gates(const float* __restrict__ x, const float* __restrict__ h,
               const float* __restrict__ cst, const float* __restrict__ agg,
               const float* __restrict__ Wi, const float* __restrict__ Wf,
               const float* __restrict__ Wc, const float* __restrict__ Wo,
               const float* __restrict__ thi, const float* __restrict__ thf,
               const float* __restrict__ thc, const float* __restrict__ tho,
               const float* __restrict__ bi, const float* __restrict__ bf,
               const float* __restrict__ bc, const float* __restrict__ bo,
               const float* __restrict__ cbi, const float* __restrict__ cbf,
               const float* __restrict__ cbc, const float* __restrict__ cbo,
               const float* __restrict__ wlin, const float* __restrict__ blin,
               float* __restrict__ out) {
  extern __shared__ float smem[];
  const int tid = threadIdx.x;

  // ---- stage all weights into LDS, transposing to K-pair-major ----
  {
    const float* Wp[4] = {Wi, Wf, Wc, Wo};
    const float* Tp[4] = {thi, thf, thc, tho};
    const float* Bp[4] = {bi, bf, bc, bo};
    const float* Cp[4] = {cbi, cbf, cbc, cbo};
    for (int i = tid; i < 4 * FIN * FOUT; i += 256) {
      int g = i >> 11, r = i & 2047;              // r = k*32 + col
      int k = r >> 5, col = r & 31;
      smem[g * 2048 + (k >> 1) * 64 + col * 2 + (k & 1)] = Wp[g][r];
    }
    for (int i = tid; i < 4 * FOUT * FOUT; i += 256) {
      int g = i >> 10, r = i & 1023;
      int k = r >> 5, col = r & 31;
      int dst = (k >> 1) * 64 + col * 2 + (k & 1);
      smem[8192 + g * 1024 + dst]  = Tp[g][r];           // theta[0]
      smem[12288 + g * 1024 + dst] = Tp[g][1024 + r];    // theta[1]
    }
    for (int i = tid; i < 4 * FOUT; i += 256) {
      smem[16384 + i] = Bp[i >> 5][i & 31];
      smem[16512 + i] = Cp[i >> 5][i & 31];
    }
    if (tid < FOUT) smem[16640 + tid] = wlin[tid];
    if (tid == 0)   smem[16672] = blin[0];
  }
  __syncthreads();

  const int lane = tid & 31;
  const int wid  = tid >> 5;
  const int base = (blockIdx.x * 8 + wid) * 16;   // first node of wave's tile
  const int m    = lane & 15;                     // A-matrix row / B,N column
  const int hi   = lane >> 4;                     // which half-wave
  const int klane = hi * 2;                       // K offset inside a K=4 chunk
  int node = base + m;
  if (node >= NODES) node = NODES - 1;            // clamp loads, stores predicated

  // ---- A-operand tiles (ISA 7.12.2: v.x = K=klane, v.y = K=klane+1) ----
  v2f xa[16];                                     // x tile: 16 x 64
#pragma unroll
  for (int c0 = 0; c0 < 16; ++c0)
    xa[c0] = *(const v2f*)(x + node * FIN + c0 * 4 + klane);

  v2f ha[8], ga[8];                               // h tile and -agg tile: 16 x 32
#pragma unroll
  for (int c0 = 0; c0 < 8; ++c0) {
    ha[c0] = *(const v2f*)(h + node * FOUT + c0 * 4 + klane);
    v2f g  = *(const v2f*)(agg + node * FOUT + c0 * 4 + klane);
    ga[c0] = -g;                                  // cheb: ... - agg @ theta[1]
  }

  // ---- pre-activations: pre[g][nb] = bias + x@W + h@th0 - agg@th1 ----
  v8f pre[4][2];
#pragma unroll
  for (int g = 0; g < 4; ++g) {
    const float* sW  = smem + g * 2048;
    const float* sT0 = smem + 8192 + g * 1024;
    const float* sT1 = smem + 12288 + g * 1024;
#pragma unroll
    for (int nb = 0; nb < 2; ++nb) {
      const int col  = nb * 16 + m;
      const int coff = col * 2;                   // K-pair-major column offset
      const float bv = smem[16384 + g * 32 + col] + smem[16512 + g * 32 + col];
      v8f acc = {bv, bv, bv, bv, bv, bv, bv, bv};
#pragma unroll
      for (int c0 = 0; c0 < 16; ++c0) {           // x @ W_g, K = 64
        v2f b = *(const v2f*)(sW + (c0 * 2 + hi) * 64 + coff);
        acc = wmma_k4(xa[c0], b, acc);
      }
#pragma unroll
      for (int c0 = 0; c0 < 8; ++c0) {            // h @ theta0, K = 32
        v2f b = *(const v2f*)(sT0 + (c0 * 2 + hi) * 64 + coff);
        acc = wmma_k4(ha[c0], b, acc);
      }
#pragma unroll
      for (int c0 = 0; c0 < 8; ++c0) {            // (-agg) @ theta1, K = 32
        v2f b = *(const v2f*)(sT1 + (c0 * 2 + hi) * 64 + coff);
        acc = wmma_k4(ga[c0], b, acc);
      }
      pre[g][nb] = acc;
    }
  }

  // ---- LSTM elementwise + ReLU + head (fused) ----
  // C/D layout: vgpr r, lanes 0-15 -> row r, lanes 16-31 -> row r+8.
  float partial[8];
#pragma unroll
  for (int r = 0; r < 8; ++r) partial[r] = 0.0f;

#pragma unroll
  for (int nb = 0; nb < 2; ++nb) {
    const int col  = nb * 16 + m;
    const float wl = smem[16640 + col];
#pragma unroll
    for (int r = 0; r < 8; ++r) {
      int nrow = base + r + 8 * hi;
      int ncl  = nrow < NODES ? nrow : NODES - 1;
      float cv = cst[ncl * FOUT + col];
      float I  = sigmoid_(pre[0][nb][r]);
      float F  = sigmoid_(pre[1][nb][r]);
      float T  = tanhf(pre[2][nb][r]);
      float O  = sigmoid_(pre[3][nb][r]);
      float Cn = F * cv + I * T;
      float Hn = O * tanhf(Cn);
      float Hr = Hn > 0.0f ? Hn : 0.0f;           // relu
      partial[r] += Hr * wl;                      // row-dot with W_lin
    }
  }

  // Reduce the 16 lanes of each half-wave (row m lives in one half).
#pragma unroll
  for (int r = 0; r < 8; ++r) {
    float v = partial[r];
    v += __shfl_xor(v, 1);
    v += __shfl_xor(v, 2);
    v += __shfl_xor(v, 4);
    v += __shfl_xor(v, 8);
    partial[r] = v;
  }

  if (m == 0) {                                   // lanes 0 and 16 write
    const float bl = smem[16672];
#pragma unroll
    for (int r = 0; r < 8; ++r) {
      int nrow = base + r + 8 * hi;
      if (nrow < NODES) out[nrow] = partial[r] + bl;
    }
  }
}

// ---------------------------------------------------------------------------
extern "C" void kernel_launch(void* const* d_in, const int* in_sizes, int n_in,
                              void* d_out, int out_size, void* d_ws, size_t ws_size,
                              hipStream_t stream) {
  const float* x    = (const float*)d_in[0];
  const int*   ei   = (const int*)d_in[1];      // [2, E]: row then col
  const float* ew   = (const float*)d_in[2];
  const float* h    = (const float*)d_in[3];
  const float* c    = (const float*)d_in[4];
  const float* Wi   = (const float*)d_in[5];
  const float* bi   = (const float*)d_in[6];
  const float* thi  = (const float*)d_in[7];
  const float* cbi  = (const float*)d_in[8];
  const float* Wf   = (const float*)d_in[9];
  const float* bf   = (const float*)d_in[10];
  const float* thf  = (const float*)d_in[11];
  const float* cbf  = (const float*)d_in[12];
  const float* Wc   = (const float*)d_in[13];
  const float* bc   = (const float*)d_in[14];
  const float* thc  = (const float*)d_in[15];
  const float* cbc  = (const float*)d_in[16];
  const float* Wo   = (const float*)d_in[17];
  const float* bo   = (const float*)d_in[18];
  const float* tho  = (const float*)d_in[19];
  const float* cbo  = (const float*)d_in[20];
  const float* wlin = (const float*)d_in[21];
  const float* blin = (const float*)d_in[22];
  float* out = (float*)d_out;

  // workspace: deg[N] (reused as d^{-1/2}) then agg[N x 32]
  const int DEG_FLOATS = ((NODES + 127) / 128) * 128;  // 100096, aligned
  float* deg = (float*)d_ws;
  float* agg = deg + DEG_FLOATS;
  hipMemsetAsync(d_ws, 0, (size_t)(DEG_FLOATS + (size_t)NODES * FOUT) * sizeof(float),
                 stream);

  const int* row = ei;
  const int* col = ei + EDGES;

  k_deg<<<(EDGES + 255) / 256, 256, 0, stream>>>(row, ew, deg);
  k_dis<<<(NODES + 255) / 256, 256, 0, stream>>>(deg);
  k_scatter<<<EDGES / 8, 256, 0, stream>>>(row, col, ew, deg, h, agg);

  const size_t smem_bytes = 16673 * sizeof(float);     // ~65 KB (LDS is 320 KB/WGP)
  k_gclstm_gates<<<(NODES + 127) / 128, 256, smem_bytes, stream>>>(
      x, h, c, agg, Wi, Wf, Wc, Wo, thi, thf, thc, tho,
      bi, bf, bc, bo, cbi, cbf, cbc, cbo, wlin, blin, out);
}